// lossFunc_51762945851682
// MI455X (gfx1250) — compile-verified
//
#include <hip/hip_runtime.h>

// ---------------------------------------------------------------------------
// DKT masked-BCE loss for MI455X (gfx1250).
//
// Memory-bound: must scan batch[:,1:50,:] (~491 MB); pred becomes a 50k-point
// gather because delta rows are one-hot with value exactly 1.0f.
// Floor: 491 MB / 23.3 TB/s ~ 21 us. WMMA would force a dense 250 MB pred
// read (+50% traffic) for a diagonal-only result -> wrong tool. Instead we
// exercise the CDNA5 async data mover: global_load_async_to_lds_b128 streams
// each row into LDS (ASYNCcnt / s_wait_asynccnt), compute reads LDS.
// ---------------------------------------------------------------------------

#define NUM_Q    1224
#define ROW_F    2448           // 2*NUM_Q floats per batch row
#define ROW_V4   612            // float4 slots per row
#define HALF_V4  306            // float4 slots in the "correct" half (1224/4)
#define SSTEPS   50
#define TPB      256            // 8 wave32 waves
#define T_PER_BLK 7             // 49 steps = 7 chunks of 7

__global__ __launch_bounds__(TPB)
void dkt_bce_partial_kernel(const float* __restrict__ pred,
                            const float* __restrict__ batch,
                            float* __restrict__ partial)
{
    __shared__ __align__(16) float row[ROW_F];   // 9792 B staging buffer
    __shared__ float red_q[8], red_c[8], red_w[8];

    const int tid  = threadIdx.x;
    const int lane = tid & 31;
    const int wid  = tid >> 5;

    const int b     = blockIdx.x / T_PER_BLK;
    const int chunk = blockIdx.x % T_PER_BLK;

    // addrspace(3) pointers live in the low 32 bits of the generic address:
    // truncation yields the workgroup-relative LDS byte offset the async op wants.
    const uint32_t lds_base = (uint32_t)(uintptr_t)(&row[0]);

    float blockSum = 0.0f;   // meaningful on thread 0 only

    for (int i = 0; i < T_PER_BLK; ++i) {
        const int t = chunk * T_PER_BLK + i;              // 0..48  (pred step)
        const float* rsrc =
            batch + ((size_t)b * SSTEPS + (size_t)(t + 1)) * (size_t)ROW_F;

        // ---- async DMA: global -> LDS, 16 B per lane per issue -------------
        #pragma unroll
        for (int k = 0; k < 3; ++k) {
            const int slot = tid + k * TPB;               // float4 slot index
            if (slot < ROW_V4) {
                unsigned long long ga = (unsigned long long)(rsrc + slot * 4);
                uint32_t la = lds_base + (uint32_t)(slot * 16);
                asm volatile("global_load_async_to_lds_b128 %0, %1, off"
                             :: "v"(la), "v"(ga) : "memory");
            }
        }
        asm volatile("s_wait_asynccnt 0" ::: "memory");   // my copies landed
        __syncthreads();                                   // everyone's landed

        // ---- one-hot extraction as a branch-free reduction -----------------
        // q*  = sum q * (f+s)   (exact integer in f32, q < 1224)
        // c   = sum (f - s)     (+1 correct / -1 incorrect)
        // w   = sum (f + s)     (mask: row actually has a question)
        float qacc = 0.0f, cacc = 0.0f, wacc = 0.0f;
        #pragma unroll
        for (int k = 0; k < 3; ++k) {
            const int slot = tid + k * TPB;
            if (slot < ROW_V4) {
                const float4 v =
                    *reinterpret_cast<const float4*>(&row[slot * 4]);
                const float e = (v.x + v.y) + (v.z + v.w);
                // each float4 is entirely inside one half (1224 = 306*4)
                const int   qb  = slot * 4 - ((slot >= HALF_V4) ? NUM_Q : 0);
                const float sgn = (slot >= HALF_V4) ? -1.0f : 1.0f;
                qacc += (float)qb * e + (v.y + 2.0f * v.z + 3.0f * v.w);
                wacc += e;
                cacc += sgn * e;
            }
        }

        // ---- deterministic block reduction (wave32 shfl + 8-way LDS) -------
        #pragma unroll
        for (int off = 16; off > 0; off >>= 1) {
            qacc += __shfl_xor(qacc, off, 32);
            cacc += __shfl_xor(cacc, off, 32);
            wacc += __shfl_xor(wacc, off, 32);
        }
        if (lane == 0) { red_q[wid] = qacc; red_c[wid] = cacc; red_w[wid] = wacc; }
        __syncthreads();

        if (tid == 0) {
            float q = 0.0f, c = 0.0f, w = 0.0f;
            #pragma unroll
            for (int j = 0; j < 8; ++j) { q += red_q[j]; c += red_c[j]; w += red_w[j]; }
            if (w > 0.5f) {                                // mask == (p > 0)
                int qstar = (int)(q + 0.5f);
                qstar = min(max(qstar, 0), NUM_Q - 1);
                const float p =
                    pred[((size_t)b * SSTEPS + (size_t)t) * (size_t)NUM_Q + qstar];
                const float a = (c > 0.5f) ? 1.0f : 0.0f;
                const float bce = a * logf(p) + (1.0f - a) * logf(1.0f - p);
                blockSum -= bce;
            }
        }
        __syncthreads();   // protect row[]/red_* before next iteration reuses them
    }

    if (tid == 0) partial[blockIdx.x] = blockSum;
}

__global__ __launch_bounds__(TPB)
void dkt_reduce_kernel(const float* __restrict__ partial,
                       float* __restrict__ out, int n)
{
    __shared__ float red[8];
    const int tid = threadIdx.x;
    float s = 0.0f;
    for (int i = tid; i < n; i += TPB) s += partial[i];   // fixed order
    #pragma unroll
    for (int off = 16; off > 0; off >>= 1) s += __shfl_xor(s, off, 32);
    if ((tid & 31) == 0) red[tid >> 5] = s;
    __syncthreads();
    if (tid == 0) {
        float t = 0.0f;
        #pragma unroll
        for (int j = 0; j < 8; ++j) t += red[j];
        out[0] = t;
    }
}

extern "C" void kernel_launch(void* const* d_in, const int* in_sizes, int n_in,
                              void* d_out, int out_size, void* d_ws, size_t ws_size,
                              hipStream_t stream)
{
    const float* pred  = (const float*)d_in[0];   // [B, 50, 1224] f32
    const float* batch = (const float*)d_in[1];   // [B, 50, 2448] f32
    float*       out   = (float*)d_out;           // [1] f32
    float*       part  = (float*)d_ws;

    const int B    = in_sizes[0] / (SSTEPS * NUM_Q);   // 1024
    const int nblk = B * T_PER_BLK;                    // 7168 partials

    dkt_bce_partial_kernel<<<nblk, TPB, 0, stream>>>(pred, batch, part);
    dkt_reduce_kernel<<<1, TPB, 0, stream>>>(part, out, nblk);
}